// TaxaNet_custom_loss_82403242541775
// MI455X (gfx1250) — compile-verified
//
#include <hip/hip_runtime.h>
#include <math.h>

// ---------------------------------------------------------------------------
// TaxaNet hierarchical loss for MI455X (gfx1250, wave32).
//   Kernel 1: one workgroup per (batch, level) row -> masked log-softmax
//             NLL + argmax. Row staged to LDS once — via
//             GLOBAL_LOAD_ASYNC_TO_LDS_B128 (ASYNCcnt) when available,
//             else b128 load + ds_store — then two LDS passes with wave32
//             shuffle + cross-wave LDS reduction.
//   Kernel 2: builds a 16 x 256 f32 matrix in LDS (rows: 4 NLL levels,
//             3 H-consistency indicator rows, rest zero) and reduces over
//             the batch with 64 chained V_WMMA_F32_16X16X4_F32 against an
//             all-ones B matrix (f32 accumulate), then emits the weighted
//             scalar loss.
// ---------------------------------------------------------------------------

typedef __attribute__((ext_vector_type(2))) float v2f;
typedef __attribute__((ext_vector_type(8))) float v8f;
typedef int bv4i __attribute__((vector_size(16)));          // 128-bit payload type
typedef __attribute__((address_space(1))) bv4i* gv4i_ptr;   // global (AS1)
typedef __attribute__((address_space(3))) bv4i* lv4i_ptr;   // LDS (AS3)

#define NLEVELS 4
#define CDIM 8000
#define SROW 260  // padded LDS stride: 260 % 64 = 4 -> 16 A-lanes hit distinct banks

#if defined(__has_builtin)
#if __has_builtin(__builtin_amdgcn_global_load_async_to_lds_b128)
#define HAVE_ASYNC_TO_LDS 1
#endif
#if __has_builtin(__builtin_amdgcn_s_wait_asynccnt)
#define HAVE_WAIT_ASYNCCNT 1
#endif
#endif

__constant__ int c_sizes[NLEVELS] = {100, 400, 2000, 8000};
__constant__ int c_offs[NLEVELS]  = {0, 100, 500, 2500};

// ---------------------------------------------------------------------------
// Kernel 1: per-row masked log-softmax NLL + argmax.
// grid = nB*4 blocks, 256 threads (8 wave32).
// ---------------------------------------------------------------------------
__global__ void __launch_bounds__(256)
row_reduce_kernel(const float* __restrict__ y_pred,
                  const int* __restrict__ y_true,
                  float* __restrict__ nll_ws,
                  int* __restrict__ g_ws)
{
    __shared__ __align__(16) float sdata[CDIM];  // 32 KB row stage
    __shared__ float swmax[8];
    __shared__ int   swidx[8];
    __shared__ float swsum[8];

    const int r   = blockIdx.x;       // row id = b*4 + j
    const int j   = r & 3;
    const int n   = c_sizes[j];
    const int tid = threadIdx.x;
    const int lane = tid & 31;
    const int wid  = tid >> 5;

    const float* row = y_pred + (size_t)r * CDIM;

    // ---- stage valid part of row into LDS (16B vectors; n is mult. of 4) ----
    const int n4 = n >> 2;
    const float4* row4 = reinterpret_cast<const float4*>(row);
    float4* sd4 = reinterpret_cast<float4*>(sdata);
#ifdef HAVE_ASYNC_TO_LDS
    // CDNA5 direct global->LDS async copy (bypasses VGPRs, tracked by ASYNCcnt)
    for (int i = tid; i < n4; i += 256) {
        __builtin_amdgcn_global_load_async_to_lds_b128(
            (gv4i_ptr)(row4 + i),
            (lv4i_ptr)(sd4 + i),
            0, 0);
    }
#ifdef HAVE_WAIT_ASYNCCNT
    __builtin_amdgcn_s_wait_asynccnt(0);
#else
    asm volatile("s_wait_asynccnt 0x0" ::: "memory");
#endif
#else
    for (int i = tid; i < n4; i += 256) sd4[i] = row4[i];
#endif
    __syncthreads();

    // ---- pass 1: max + argmax (first-occurrence tie-break) ----
    float vmax = -INFINITY;
    int   imax = 0x7fffffff;
    for (int i = tid; i < n; i += 256) {
        float v = sdata[i];
        if (v > vmax) { vmax = v; imax = i; }
    }
    for (int off = 16; off > 0; off >>= 1) {
        float ov = __shfl_down(vmax, off);
        int   oi = __shfl_down(imax, off);
        if (ov > vmax || (ov == vmax && oi < imax)) { vmax = ov; imax = oi; }
    }
    if (lane == 0) { swmax[wid] = vmax; swidx[wid] = imax; }
    __syncthreads();
    if (tid == 0) {
        for (int w = 1; w < 8; ++w) {
            float ov = swmax[w]; int oi = swidx[w];
            if (ov > vmax || (ov == vmax && oi < imax)) { vmax = ov; imax = oi; }
        }
        swmax[0] = vmax; swidx[0] = imax;
    }
    __syncthreads();
    const float rowmax = swmax[0];

    // ---- pass 2: sum of exp(x - rowmax) ----
    float s = 0.0f;
    for (int i = tid; i < n; i += 256) s += expf(sdata[i] - rowmax);
    for (int off = 16; off > 0; off >>= 1) s += __shfl_down(s, off);
    if (lane == 0) swsum[wid] = s;
    __syncthreads();

    if (tid == 0) {
        float tot = 0.0f;
        for (int w = 0; w < 8; ++w) tot += swsum[w];
        const int t  = y_true[r];           // target index (< n, valid)
        const float xt = sdata[t];
        nll_ws[r] = logf(tot) + rowmax - xt;   // -log_softmax at target
        g_ws[r]   = swidx[0] + c_offs[j];      // global node id of argmax
    }
}

// ---------------------------------------------------------------------------
// Kernel 2: batch reduction via WMMA f32 16x16x4 + final weighted combine.
// 1 block, 256 threads (column tid owns batch element tid).
// ---------------------------------------------------------------------------
__global__ void __launch_bounds__(256)
combine_kernel(const float* __restrict__ nll_ws,
               const int* __restrict__ g_ws,
               const float* __restrict__ H,
               float* __restrict__ out,
               int nB)
{
    __shared__ float S[16 * SROW];  // 16 rows x 256 cols (stride 260)

    const int tid = threadIdx.x;

    // zero my column in all 16 rows, then fill rows 0..6 for valid batch cols
    for (int m = 0; m < 16; ++m) S[m * SROW + tid] = 0.0f;
    if (tid < nB) {
        const int b4 = tid * 4;
        for (int j = 0; j < 4; ++j) S[j * SROW + tid] = nll_ws[b4 + j];
        const int g0 = g_ws[b4 + 0];
        const int g1 = g_ws[b4 + 1];
        const int g2 = g_ws[b4 + 2];
        const int g3 = g_ws[b4 + 3];
        S[4 * SROW + tid] = (H[(size_t)g0 * 10500 + g1] == 1.0f) ? 1.0f : 0.0f;
        S[5 * SROW + tid] = (H[(size_t)g1 * 10500 + g2] == 1.0f) ? 1.0f : 0.0f;
        S[6 * SROW + tid] = (H[(size_t)g2 * 10500 + g3] == 1.0f) ? 1.0f : 0.0f;
    }
    __syncthreads();

    if (tid < 32) {  // wave 0 only; EXEC all-1s within this wave at runtime
        v8f acc = {};
        v2f bones; bones.x = 1.0f; bones.y = 1.0f;  // B = ones(4x16): D[m,n] = sum_k A[m,k]
        const int m    = tid & 15;
        const int koff = (tid >> 4) ? 2 : 0;  // lanes 16..31 carry K=2,3
        #pragma unroll 4
        for (int kc = 0; kc < 64; ++kc) {     // 64 * K4 = 256 batch columns
            const int col = kc * 4 + koff;
            v2f a;
            a.x = S[m * SROW + col];
            a.y = S[m * SROW + col + 1];
            acc = __builtin_amdgcn_wmma_f32_16x16x4_f32(
                false, a, false, bones, (short)0, acc, false, false);
        }
        if (tid == 0) {
            // lane 0, VGPR m <-> D[M=m, N=0] = row-m batch sum
            const float invB = 1.0f / (float)nB;
            const float ce1 = acc[1] * invB;
            const float ce2 = acc[2] * invB;
            const float ce3 = acc[3] * invB;
            const float E = 2.71828182845904523536f;
            const float cons1 = E * acc[4];
            const float cons2 = E * acc[5];
            const float cons3 = E * acc[6];
            out[0] = 0.25f * (cons1 + ce1)
                   + 0.15f * (cons2 + ce2)
                   + 0.10f * (cons3 + ce3);
        }
    }
}

// ---------------------------------------------------------------------------
extern "C" void kernel_launch(void* const* d_in, const int* in_sizes, int n_in,
                              void* d_out, int out_size, void* d_ws, size_t ws_size,
                              hipStream_t stream)
{
    const float* y_pred = (const float*)d_in[0];   // (B, 4, 8000) f32
    const int*   y_true = (const int*)d_in[1];     // (B, 4) int
    const float* H      = (const float*)d_in[2];   // (10500, 10500) f32
    float* out = (float*)d_out;

    const int nB = in_sizes[0] / (NLEVELS * CDIM); // 256
    const int nrows = nB * NLEVELS;                // 1024

    float* nll_ws = (float*)d_ws;
    int*   g_ws   = (int*)(nll_ws + nrows);

    hipLaunchKernelGGL(row_reduce_kernel, dim3(nrows), dim3(256), 0, stream,
                       y_pred, y_true, nll_ws, g_ws);
    hipLaunchKernelGGL(combine_kernel, dim3(1), dim3(256), 0, stream,
                       nll_ws, g_ws, H, out, nB);
}